// OptimizedDistance_52561809768954
// MI455X (gfx1250) — compile-verified
//
#include <hip/hip_runtime.h>

// ---------------------------------------------------------------------------
// Neighbor list (brute force, batch-restricted) for MI455X / gfx1250.
// d2 tiles computed with V_WMMA_F32_16X16X4_F32:
//   A row i (16x4):  (-2x_i, -2y_i, -2z_i, 1)
//   B col j (4x16):  ( x_j,   y_j,   z_j,  |p_j|^2 )
//   C row i       :  |p_i|^2  broadcast
//   D = C + A*B   =  |p_i|^2 + |p_j|^2 - 2 p_i.p_j = d2(i,j)
// Row-major compaction via count / scan / ordered-fill passes.
// Hot loop: zero data-dependent branches (diagonal chunk split out of the
// loop; unconditional clamped prefetch), ballots fold to v_cmp masks and
// stay scalar (readfirstlane), SALU popcount accumulation.
// ---------------------------------------------------------------------------

typedef float v2f __attribute__((ext_vector_type(2)));
typedef float v8f __attribute__((ext_vector_type(8)));

#define WAVES_PER_BLOCK 4
#define CUT2 25.0f

// diagonal-conflict lanes for row pair (v, v+8): ballot bits v and 24+v
#define DIAG_MASK(v) (~((1u << (v)) | (1u << (24 + (v)))))

// ---------------- init: padding values for the whole output ----------------
__global__ void nl_init_out(float* __restrict__ out, int out_size, int maxPairs) {
  int idx = blockIdx.x * blockDim.x + threadIdx.x;
  int stride = gridDim.x * blockDim.x;
  for (int k = idx; k < out_size; k += stride)
    out[k] = (k < 2 * maxPairs) ? -1.0f : 0.0f;
}

// ---------------- batch ranges (batch is sorted) ---------------------------
__global__ void nl_batch_bounds(const int* __restrict__ batch, int n,
                                int* __restrict__ bStart, int* __restrict__ bEnd) {
  int i = blockIdx.x * blockDim.x + threadIdx.x;
  if (i >= n) return;
  int b = batch[i];
  if (i == 0     || batch[i - 1] != b) bStart[b] = i;
  if (i == n - 1 || batch[i + 1] != b) bEnd[b]   = i + 1;
}

// ---------------- pass 1: per-row valid-pair counts ------------------------
__global__ void __launch_bounds__(32 * WAVES_PER_BLOCK)
nl_count(const float* __restrict__ pos, const int* __restrict__ batch, int n,
         const int* __restrict__ bStart, const int* __restrict__ bEnd,
         int* __restrict__ rowCount) {
  const int lane   = threadIdx.x & 31;
  const int wave   = threadIdx.x >> 5;
  const int stripe = blockIdx.x * WAVES_PER_BLOCK + wave;
  const int i0     = stripe * 16;
  if (i0 >= n) return;                 // uniform per wave -> EXEC stays full
  const int half = lane >> 4;
  const int lp   = lane & 15;

  // A operand: lanes 0-15 hold (K0,K1)=(-2x,-2y), lanes 16-31 hold (K2,K3)=(-2z,1)
  float ax, ay;
  {
    int r = i0 + lp;
    float x = pos[3 * r + 0], y = pos[3 * r + 1], z = pos[3 * r + 2];
    if (half == 0) { ax = -2.0f * x; ay = -2.0f * y; }
    else           { ax = -2.0f * z; ay = 1.0f;      }
  }
  // Per-VGPR row data: VGPR v maps to row i0 + v + 8*half
  float sqi[8]; int bi[8];
#pragma unroll
  for (int v = 0; v < 8; ++v) {
    int m = i0 + v + 8 * half;
    float x = pos[3 * m + 0], y = pos[3 * m + 1], z = pos[3 * m + 2];
    sqi[v] = x * x + y * y + z * z;
    bi[v]  = batch[m];
  }

  // scalarize loop bounds (uniform-address loads look divergent to LLVM)
  const int jLo = __builtin_amdgcn_readfirstlane(bStart[batch[i0]]);
  const int jHi = __builtin_amdgcn_readfirstlane(bEnd[batch[i0 + 15]]);
  const int jc0 = (jLo >> 4) << 4;
  const int lastChunk = ((jHi - 1) >> 4) << 4;

  int accLo[8], accHi[8];              // scalar (SALU) accumulators
#pragma unroll
  for (int v = 0; v < 8; ++v) { accLo[v] = 0; accHi[v] = 0; }

  // pipelined column state
  float xj, yj, zj; int bj;
  {
    int j = jc0 + lp;
    xj = pos[3 * j + 0]; yj = pos[3 * j + 1]; zj = pos[3 * j + 2];
    bj = batch[j];
  }

  auto chunk = [&](int jc, int jn, bool isDiag) __attribute__((always_inline)) {
    float cx = xj, cy = yj, cz = zj;
    int   cb = bj;
    (void)jc;
    {                                  // unconditional clamped prefetch
      int j = jn + lp;
      xj = pos[3 * j + 0]; yj = pos[3 * j + 1]; zj = pos[3 * j + 2];
      bj = batch[j];
    }
    float sqj = cx * cx + cy * cy + cz * cz;
    v2f a; a.x = ax; a.y = ay;
    v2f b; b.x = half ? cz : cx; b.y = half ? sqj : cy;
    v8f c;
#pragma unroll
    for (int v = 0; v < 8; ++v) c[v] = sqi[v];
    c = __builtin_amdgcn_wmma_f32_16x16x4_f32(false, a, false, b,
                                              (short)0, c, false, false);
#pragma unroll
    for (int v = 0; v < 8; ++v) {
      // max(c,0) < 25  <=>  c < 25  (incl. NaN -> false), so no clamp needed
      unsigned m = __builtin_amdgcn_ballot_w32(c[v] < CUT2) &
                   __builtin_amdgcn_ballot_w32(bi[v] == cb);
      unsigned bm = (unsigned)__builtin_amdgcn_readfirstlane((int)m);
      if (isDiag) bm &= DIAG_MASK(v);  // folds to constant at the one call
      accLo[v] += __builtin_popcount(bm & 0xFFFFu);
      accHi[v] += __builtin_popcount(bm >> 16);
    }
  };

  for (int jc = jc0; jc < i0; jc += 16)        // pre-diagonal chunks
    chunk(jc, jc + 16, false);
  chunk(i0, min(i0 + 16, lastChunk), true);    // the diagonal chunk
  for (int jc = i0 + 16; jc < jHi; jc += 16)   // post-diagonal chunks
    chunk(jc, min(jc + 16, lastChunk), false);

  if (lane == 0) {
#pragma unroll
    for (int v = 0; v < 8; ++v) {
      rowCount[i0 + v]     = accLo[v];
      rowCount[i0 + 8 + v] = accHi[v];
    }
  }
}

// ---------------- pass 2: exclusive scan over row counts (1 block) ---------
__global__ void __launch_bounds__(1024)
nl_scan(const int* __restrict__ rowCount, int n, int* __restrict__ rowOff) {
  __shared__ int s[1024];
  int t = threadIdx.x;
  int per = n >> 10;                 // rows per thread (16 for N=16384)
  if (per > 16) per = 16;
  int base = t * per;
  int local[16];
  int sum = 0;
  for (int k = 0; k < per; ++k) { local[k] = rowCount[base + k]; sum += local[k]; }
  s[t] = sum;
  __syncthreads();
  for (int off = 1; off < 1024; off <<= 1) {
    int v = (t >= off) ? s[t - off] : 0;
    __syncthreads();
    s[t] += v;
    __syncthreads();
  }
  int run = s[t] - sum;              // exclusive base for this thread's rows
  for (int k = 0; k < per; ++k) { rowOff[base + k] = run; run += local[k]; }
}

// ---------------- pass 3: ordered fill -------------------------------------
__global__ void __launch_bounds__(32 * WAVES_PER_BLOCK)
nl_fill(const float* __restrict__ pos, const int* __restrict__ batch, int n,
        const int* __restrict__ bStart, const int* __restrict__ bEnd,
        const int* __restrict__ rowOff, int maxPairs,
        float* __restrict__ outI, float* __restrict__ outJ,
        float* __restrict__ outW, float* __restrict__ outV) {
  const int lane   = threadIdx.x & 31;
  const int wave   = threadIdx.x >> 5;
  const int stripe = blockIdx.x * WAVES_PER_BLOCK + wave;
  const int i0     = stripe * 16;
  if (i0 >= n) return;
  const int half = lane >> 4;
  const int lp   = lane & 15;

  float ax, ay;
  {
    int r = i0 + lp;
    float x = pos[3 * r + 0], y = pos[3 * r + 1], z = pos[3 * r + 2];
    if (half == 0) { ax = -2.0f * x; ay = -2.0f * y; }
    else           { ax = -2.0f * z; ay = 1.0f;      }
  }
  float sqi[8], pix[8], piy[8], piz[8];
  int bi[8], rowId[8], baseOff[8];
  int wrLo[8], wrHi[8];                // scalar (SALU) write counters
#pragma unroll
  for (int v = 0; v < 8; ++v) {
    int m = i0 + v + 8 * half;
    float x = pos[3 * m + 0], y = pos[3 * m + 1], z = pos[3 * m + 2];
    pix[v] = x; piy[v] = y; piz[v] = z;
    sqi[v]   = x * x + y * y + z * z;
    bi[v]    = batch[m];
    rowId[v] = m;
    baseOff[v] = rowOff[m];
    wrLo[v] = 0; wrHi[v] = 0;
  }

  const int jLo = __builtin_amdgcn_readfirstlane(bStart[batch[i0]]);
  const int jHi = __builtin_amdgcn_readfirstlane(bEnd[batch[i0 + 15]]);
  const int jc0 = (jLo >> 4) << 4;
  const int lastChunk = ((jHi - 1) >> 4) << 4;

  float xj, yj, zj; int bj;
  {
    int j = jc0 + lp;
    xj = pos[3 * j + 0]; yj = pos[3 * j + 1]; zj = pos[3 * j + 2];
    bj = batch[j];
  }

  auto chunk = [&](int jc, int jn, bool isDiag) __attribute__((always_inline)) {
    float cx = xj, cy = yj, cz = zj;
    int   cb = bj;
    {                                  // unconditional clamped prefetch
      int j = jn + lp;
      xj = pos[3 * j + 0]; yj = pos[3 * j + 1]; zj = pos[3 * j + 2];
      bj = batch[j];
    }
    float sqj = cx * cx + cy * cy + cz * cz;
    int   jg  = jc + lp;
    v2f a; a.x = ax; a.y = ay;
    v2f b; b.x = half ? cz : cx; b.y = half ? sqj : cy;
    v8f c;
#pragma unroll
    for (int v = 0; v < 8; ++v) c[v] = sqi[v];
    c = __builtin_amdgcn_wmma_f32_16x16x4_f32(false, a, false, b,
                                              (short)0, c, false, false);
#pragma unroll
    for (int v = 0; v < 8; ++v) {
      unsigned m = __builtin_amdgcn_ballot_w32(c[v] < CUT2) &
                   __builtin_amdgcn_ballot_w32(bi[v] == cb);
      unsigned bm = (unsigned)__builtin_amdgcn_readfirstlane((int)m);
      if (isDiag) bm &= DIAG_MASK(v);
      if ((bm >> lane) & 1u) {         // divergent only around the stores
        unsigned my = half ? (bm >> 16) : (bm & 0xFFFFu);
        int wr   = half ? wrHi[v] : wrLo[v];
        int rank = __builtin_popcount(my & ((1u << lp) - 1u));
        int p = baseOff[v] + wr + rank;
        if (p < maxPairs) {
          float dx = pix[v] - cx, dy = piy[v] - cy, dz = piz[v] - cz;
          outI[p] = (float)rowId[v];
          outJ[p] = (float)jg;
          outW[p] = sqrtf(dx * dx + dy * dy + dz * dz);
          outV[3 * p + 0] = dx;
          outV[3 * p + 1] = dy;
          outV[3 * p + 2] = dz;
        }
      }
      wrLo[v] += __builtin_popcount(bm & 0xFFFFu);
      wrHi[v] += __builtin_popcount(bm >> 16);
    }
  };

  for (int jc = jc0; jc < i0; jc += 16)        // pre-diagonal chunks
    chunk(jc, jc + 16, false);
  chunk(i0, min(i0 + 16, lastChunk), true);    // the diagonal chunk
  for (int jc = i0 + 16; jc < jHi; jc += 16)   // post-diagonal chunks
    chunk(jc, min(jc + 16, lastChunk), false);
}

// ---------------------------------------------------------------------------
extern "C" void kernel_launch(void* const* d_in, const int* in_sizes, int n_in,
                              void* d_out, int out_size, void* d_ws, size_t ws_size,
                              hipStream_t stream) {
  const float* pos   = (const float*)d_in[0];
  const int*   batch = (const int*)d_in[1];
  const int n        = in_sizes[1];        // N atoms
  const int maxPairs = 64 * n;

  int* ws        = (int*)d_ws;
  int* rowCount  = ws;                     // n ints
  int* rowOff    = ws + n;                 // n ints
  int* bStart    = ws + 2 * n;             // 16 ints
  int* bEnd      = ws + 2 * n + 16;        // 16 ints

  float* outF = (float*)d_out;
  float* outI = outF;
  float* outJ = outF + maxPairs;
  float* outW = outF + 2 * maxPairs;
  float* outV = outF + 3 * maxPairs;

  // init padding (edge_index = -1, w/vec = 0) for the whole output buffer
  nl_init_out<<<2048, 256, 0, stream>>>(outF, out_size, maxPairs);

  // batch start/end indices (batch is sorted)
  nl_batch_bounds<<<(n + 255) / 256, 256, 0, stream>>>(batch, n, bStart, bEnd);

  const int stripes = n / 16;
  const int blocks  = (stripes + WAVES_PER_BLOCK - 1) / WAVES_PER_BLOCK;

  nl_count<<<blocks, 32 * WAVES_PER_BLOCK, 0, stream>>>(
      pos, batch, n, bStart, bEnd, rowCount);

  nl_scan<<<1, 1024, 0, stream>>>(rowCount, n, rowOff);

  nl_fill<<<blocks, 32 * WAVES_PER_BLOCK, 0, stream>>>(
      pos, batch, n, bStart, bEnd, rowOff, maxPairs,
      outI, outJ, outW, outV);
}